// Standard_NeuralDecoder_88802743812480
// MI455X (gfx1250) — compile-verified
//
#include <hip/hip_runtime.h>
#include <stdint.h>

#define NNODES 16384
#define NVARS  12288
#define BATCH  64
#define EBASE  196608
#define NITERS 10

typedef __attribute__((ext_vector_type(2))) float v2f;
typedef __attribute__((ext_vector_type(8))) float v8f;

// ---------------- CDNA5 async global->LDS helpers (inline asm, gfx1250) -----
// GV addressing mode: 64-bit per-lane address in a VGPR pair, SADDR = off.
__device__ __forceinline__ void async_ld_i32(void* lds_ptr, const int* gaddr) {
  unsigned l = (unsigned)(uintptr_t)lds_ptr;   // generic LDS ptr low 32 = offset
  unsigned long long a = (unsigned long long)(uintptr_t)gaddr;
  asm volatile("global_load_async_to_lds_b32 %0, %1, off"
               :: "v"(l), "v"(a)
               : "memory");
}
__device__ __forceinline__ void wait_async0() {
  asm volatile("s_wait_asynccnt 0x0" ::: "memory");
}
__device__ __forceinline__ int rfl(int v) {            // pin wave-uniform -> SGPR
  return __builtin_amdgcn_readfirstlane(v);
}

// branchless fixed-depth search: largest d in [0,32) with rowLds[d]-R0 <= p
__device__ __forceinline__ int rowOfSlot(const int* rowLds, int R0, int p) {
  int lo = 0, hi = 32;
#pragma unroll
  for (int i = 0; i < 5; ++i) {
    int mid = (lo + hi) >> 1;
    bool ge = (rowLds[mid] - R0 <= p);
    lo = ge ? mid : lo;
    hi = ge ? hi : mid;
  }
  return lo;
}

// ---------------- 1) build base_t / x0 in [node][batch] layout --------------
__global__ void __launch_bounds__(256) k_init(const float* __restrict__ llr,
                                              float* __restrict__ baseT,
                                              float* __restrict__ x0) {
  __shared__ float tile[64][65];
  const int t = threadIdx.x;
  const int n0 = blockIdx.x * 64;
  const int c = t & 63, q = t >> 6;
  for (int i = 0; i < 16; ++i) {        // load llr[b][n0+c], coalesced over c
    int b = i * 4 + q;
    int n = n0 + c;
    tile[b][c] = (n < NVARS) ? llr[(size_t)b * NVARS + n] : 0.0f;
  }
  __syncthreads();
  for (int i = 0; i < 16; ++i) {        // baseT[(n0+nl)*64 + c] = tile[c][nl]
    int nl = i * 4 + q;
    float v = tile[c][nl];
    size_t idx = (size_t)(n0 + nl) * 64 + c;
    baseT[idx] = v;
    x0[idx] = v;
  }
}

// ---------------- 2) CSR build ----------------------------------------------
__global__ void k_zero(int* __restrict__ p, int n) {
  int i = blockIdx.x * blockDim.x + threadIdx.x;
  if (i < n) p[i] = 0;
}

__global__ void k_hist(const int* __restrict__ dstE, int* __restrict__ deg) {
  for (int e = blockIdx.x * blockDim.x + threadIdx.x; e < EBASE;
       e += gridDim.x * blockDim.x)
    atomicAdd(&deg[dstE[e]], 1);
}

__global__ void __launch_bounds__(1024) k_scan(int* __restrict__ degcur,
                                               int* __restrict__ rowStart) {
  __shared__ int part[1024];
  const int tid = threadIdx.x;
  const int base = tid * 16;
  int sum = 0;
  for (int j = 0; j < 16; ++j) sum += degcur[base + j];
  part[tid] = sum;
  __syncthreads();
  for (int off = 1; off < 1024; off <<= 1) {
    int v = (tid >= off) ? part[tid - off] : 0;
    __syncthreads();
    part[tid] += v;
    __syncthreads();
  }
  int run = (tid == 0) ? 0 : part[tid - 1];
  for (int j = 0; j < 16; ++j) {
    int n = base + j;
    int d = degcur[n];
    rowStart[n] = run;
    degcur[n] = run;          // becomes fill cursor
    run += d;
  }
  if (tid == 1023) rowStart[NNODES] = run;   // == EBASE
}

__global__ void k_fill(const int* __restrict__ srcE, const int* __restrict__ dstE,
                       int* __restrict__ cursor, int* __restrict__ csr) {
  for (int e = blockIdx.x * blockDim.x + threadIdx.x; e < EBASE;
       e += gridDim.x * blockDim.x) {
    int slot = atomicAdd(&cursor[dstE[e]], 1);
    csr[slot] = srcE[e];
  }
}

// ---------------- 3) WMMA iteration: x' = base + w * A x --------------------
#define GCAP 2048

template <bool STAGED>
__device__ __forceinline__ v8f gather_loop(
    const int* __restrict__ csr, int R0,
    const int* eidx, const int* slot2row, const int* rowLds,
    int gS, int gE, int mg, int h, const float* __restrict__ xc, float w) {
  v8f acc = {0.f, 0.f, 0.f, 0.f, 0.f, 0.f, 0.f, 0.f};
  const int nCh = ((gE - gS) + 3) >> 2;          // scalar: gS/gE are SGPRs
  for (int c = 0; c < nCh; ++c) {
    // this lane's two K slots: K = {2h, 2h+1}
    int s0 = gS + c * 4 + 2 * h;
    int s1 = s0 + 1;
    bool ok0 = s0 < gE, ok1 = s1 < gE;
    int sc0 = ok0 ? s0 : gE - 1;           // clamp (loop ran => gE > gS >= 0)
    int sc1 = ok1 ? s1 : gE - 1;
    int e0, e1, r0, r1;
    if (STAGED) {
      e0 = eidx[sc0]; e1 = eidx[sc1];       // LDS broadcast reads
      r0 = slot2row[sc0]; r1 = slot2row[sc1];
    } else {
      e0 = csr[R0 + sc0]; e1 = csr[R0 + sc1];
      r0 = rowOfSlot(rowLds, R0, sc0); r1 = rowOfSlot(rowLds, R0, sc1);
    }
    v2f a, b;
    a.x = (ok0 && r0 == mg) ? w : 0.0f;    // A[M=mn][K=2h]   (w folded in)
    a.y = (ok1 && r1 == mg) ? w : 0.0f;    // A[M=mn][K=2h+1]
    b.x = xc[(size_t)e0 * 64];             // B[K=2h]  [N=mn]
    b.y = xc[(size_t)e1 * 64];             // B[K=2h+1][N=mn]
    acc = __builtin_amdgcn_wmma_f32_16x16x4_f32(false, a, false, b,
                                                (short)0, acc, false, false);
  }
  return acc;
}

__global__ void __launch_bounds__(256) k_gather(
    const float* __restrict__ xCur, float* __restrict__ xNext,
    const float* __restrict__ baseT, const int* __restrict__ rowStart,
    const int* __restrict__ csr, const float* __restrict__ wgt, int it) {
  __shared__ int eidx[GCAP];
  __shared__ int slot2row[GCAP];
  __shared__ int rowLds[33];
  const int t = threadIdx.x;
  const int wave = t >> 5, lane = t & 31;
  const int dBase = blockIdx.x * 32;

  if (t < 33) rowLds[t] = rowStart[dBase + t];
  __syncthreads();
  const int R0  = rfl(rowLds[0]);            // block-uniform -> SGPR
  const int cnt = rfl(rowLds[32]) - R0;
  const bool staged = (cnt <= GCAP);         // scalar branch
  if (staged) {
    const int* gbase = csr + R0;
    for (int p = t; p < cnt; p += 256)
      async_ld_i32(&eidx[p], gbase + p);     // GV mode: per-lane 64-bit addr
    // overlap with the async loads: build slot -> block-local row table
    for (int p = t; p < cnt; p += 256)
      slot2row[p] = rowOfSlot(rowLds, R0, p);
  }
  wait_async0();          // each wave waits for its own async loads
  __syncthreads();        // all waves' LDS writes visible

  const int g = wave >> 2;                   // dst group (16 nodes)
  const int colBase = (wave & 3) * 16;       // 16-wide batch column block
  const int gS = rfl(rowLds[g * 16])      - R0;  // wave-uniform -> SGPR
  const int gE = rfl(rowLds[g * 16 + 16]) - R0;
  const int h  = lane >> 4;                  // lane half: selects K pair
  const int mn = lane & 15;                  // M (for A) / N (for B)
  const int mg = g * 16 + mn;                // block-local row this lane owns
  const float w = wgt[it];
  const float* xc = xCur + colBase + mn;     // per-lane gather base

  v8f acc;
  if (staged)
    acc = gather_loop<true >(csr, R0, eidx, slot2row, rowLds,
                             gS, gE, mg, h, xc, w);
  else
    acc = gather_loop<false>(csr, R0, eidx, slot2row, rowLds,
                             gS, gE, mg, h, xc, w);

  // D layout: VGPR r -> lanes 0-15: (M=r, N=lane), lanes 16-31: (M=r+8)
  const int dG = dBase + g * 16;
#pragma unroll
  for (int r = 0; r < 8; ++r) {
    int node = dG + r + 8 * h;
    size_t idx = (size_t)node * 64 + colBase + mn;
    xNext[idx] = acc[r] + baseT[idx];
  }
}

// ---------------- 4) emit out[it][b][v] = x[v][b] (tiled transpose) ---------
__global__ void __launch_bounds__(256) k_emit(const float* __restrict__ xT,
                                              float* __restrict__ out) {
  __shared__ float tile[64][65];
  const int t = threadIdx.x;
  const int n0 = blockIdx.x * 64;
  const int c = t & 63, q = t >> 6;
  for (int i = 0; i < 16; ++i) {
    int n = i * 4 + q;
    tile[n][c] = xT[(size_t)(n0 + n) * 64 + c];   // coalesced over batch c
  }
  __syncthreads();
  for (int i = 0; i < 16; ++i) {
    int b = i * 4 + q;
    out[(size_t)b * NVARS + n0 + c] = tile[c][b]; // coalesced over var c
  }
}

// ---------------- launch ----------------------------------------------------
extern "C" void kernel_launch(void* const* d_in, const int* in_sizes, int n_in,
                              void* d_out, int out_size, void* d_ws, size_t ws_size,
                              hipStream_t stream) {
  const float* llr = (const float*)d_in[0];
  const int*   ei  = (const int*)d_in[1];
  const float* wgt = (const float*)d_in[2];
  const int* srcE = ei;                             // batch-0 base edges
  const int* dstE = ei + (size_t)BATCH * EBASE;
  float* out = (float*)d_out;

  char* ws = (char*)d_ws;
  float* xA       = (float*)(ws);                                   // 4 MiB
  float* xB       = (float*)(ws + ((size_t)4 << 20));               // 4 MiB
  float* baseT    = (float*)(ws + ((size_t)8 << 20));               // 4 MiB
  int*   rowStart = (int*)  (ws + ((size_t)12 << 20));              // 128 KiB
  int*   cursor   = (int*)  (ws + ((size_t)12 << 20) + (128 << 10));// 64 KiB
  int*   csr      = (int*)  (ws + ((size_t)12 << 20) + (256 << 10));// 768 KiB
  (void)in_sizes; (void)n_in; (void)out_size; (void)ws_size;

  k_init<<<NNODES / 64, 256, 0, stream>>>(llr, baseT, xA);
  k_zero<<<(NNODES + 255) / 256, 256, 0, stream>>>(cursor, NNODES);
  k_hist<<<512, 256, 0, stream>>>(dstE, cursor);
  k_scan<<<1, 1024, 0, stream>>>(cursor, rowStart);
  k_fill<<<512, 256, 0, stream>>>(srcE, dstE, cursor, csr);

  float* cur = xA;
  float* nxt = xB;
  for (int it = 0; it < NITERS; ++it) {
    k_gather<<<NNODES / 32, 256, 0, stream>>>(cur, nxt, baseT, rowStart, csr, wgt, it);
    k_emit<<<NVARS / 64, 256, 0, stream>>>(nxt, out + (size_t)it * BATCH * NVARS);
    float* tmp = cur; cur = nxt; nxt = tmp;
  }
}